// Laplacian_22711787061657
// MI455X (gfx1250) — compile-verified
//
#include <hip/hip_runtime.h>

// Periodic 2nd-difference stencil (Laplacian-like), HBM-bandwidth-bound.
// x: (32, 1, 1024, 1024) f32 -> out same shape.
//
// Roofline: 128 MiB in + 128 MiB out, ~1.25 flop/byte -> ~11 us floor at
// 23.3 TB/s. Strategy:
//   * 8 output rows per block: 10 column b128 loads + 8 b64 left-neighbor
//     loads produce 8 b128 outputs (register reuse of the vertical stencil;
//     contiguous rows fold into immediate offsets off one base address).
//   * Shared first-differences: 9 vertical + 5-per-row horizontal first
//     diffs feed the second diffs, bit-identical op order to the reference
//     nested rolls.
//   * NT stores keep the 128 MB write stream out of the 192 MB L2 so the
//     input (rows re-read by neighboring row-blocks) stays L2-resident.

namespace {

constexpr int kH = 1024;
constexpr int kW = 1024;
constexpr int kRows = 8;                        // output rows per block

typedef float v4f __attribute__((ext_vector_type(4)));  // native vector: NT-store legal

__device__ __forceinline__ float scrub1(float v) {  // remove_nan(x, 1.0)
    return (v != v) ? 1.0f : v;
}

__global__ __launch_bounds__(256)
void laplacian_rows8_kernel(const float* __restrict__ x, float* __restrict__ out) {
    const int blk = (int)blockIdx.x;
    const int img = blk >> 7;                   // 128 blocks per 1024-row image
    const int ib  = (blk & 127) << 3;           // first output row: 0..1016
    const size_t base = (size_t)img * (size_t)kH * (size_t)kW;

    const int j  = (int)threadIdx.x << 2;       // column base: 0..1020 step 4
    const int jp = (j + kW - 4) & (kW - 1);     // wrapped left-neighbor vector

    // Rows ib-2 .. ib+7 (the two above wrap; ib..ib+7 are contiguous).
    int rows[kRows + 2];
    rows[0] = (ib + kH - 2) & (kH - 1);
    rows[1] = (ib + kH - 1) & (kH - 1);
#pragma unroll
    for (int t = 0; t < kRows; ++t) rows[t + 2] = ib + t;

    // Column vectors x[rows[s], j..j+3], scrubbed (remove_nan -> 1.0).
    float c[kRows + 2][4];
#pragma unroll
    for (int s = 0; s < kRows + 2; ++s) {
        const v4f v = *reinterpret_cast<const v4f*>(x + base + (size_t)rows[s] * kW + j);
        c[s][0] = scrub1(v.x); c[s][1] = scrub1(v.y);
        c[s][2] = scrub1(v.z); c[s][3] = scrub1(v.w);
    }

    // Left neighbors x[r, j-2], x[r, j-1] for the output rows
    // (compiler trims these to b64 loads of the .z/.w halves).
    float lz[kRows], lw[kRows];
#pragma unroll
    for (int t = 0; t < kRows; ++t) {
        const v4f p = *reinterpret_cast<const v4f*>(x + base + (size_t)rows[t + 2] * kW + jp);
        lz[t] = scrub1(p.z);
        lw[t] = scrub1(p.w);
    }

    // Vertical first diffs: g1y[s] = g1_H at image row rows[s+1]
    //                               = x[rows[s]] - x[rows[s+1]]   (roll(x,1)-x)
    float g1y[kRows + 1][4];
#pragma unroll
    for (int s = 0; s < kRows + 1; ++s) {
#pragma unroll
        for (int k = 0; k < 4; ++k) {
            g1y[s][k] = c[s][k] - c[s + 1][k];
        }
    }

#pragma unroll
    for (int t = 0; t < kRows; ++t) {           // output row rows[t+2]
        const float* cc = c[t + 2];

        // Horizontal first diffs g1_W at cols j-1 .. j+3 (x[c-1]-x[c]).
        float g1x[5];
        g1x[0] = lz[t] - lw[t];
        g1x[1] = lw[t] - cc[0];
        g1x[2] = cc[0] - cc[1];
        g1x[3] = cc[1] - cc[2];
        g1x[4] = cc[2] - cc[3];

        float r[4];
#pragma unroll
        for (int k = 0; k < 4; ++k) {
            const float gx = g1y[t][k] - g1y[t + 1][k];  // 2nd diff along H
            const float gy = g1x[k] - g1x[k + 1];        // 2nd diff along W
            float s2 = gx + gy;
            s2 = fminf(fmaxf(s2, -1.0f), 1.0f) * 0.5f;   // clip then *0.5
            r[k] = s2;
        }

        v4f o;
        o.x = r[0]; o.y = r[1]; o.z = r[2]; o.w = r[3];
        // Non-temporal 128-bit store: keep write stream out of L2.
        __builtin_nontemporal_store(
            o, reinterpret_cast<v4f*>(out + base + (size_t)rows[t + 2] * kW + j));
    }
}

}  // namespace

extern "C" void kernel_launch(void* const* d_in, const int* in_sizes, int n_in,
                              void* d_out, int out_size, void* d_ws, size_t ws_size,
                              hipStream_t stream) {
    (void)n_in; (void)out_size; (void)d_ws; (void)ws_size;
    const float* x = (const float*)d_in[0];
    float* out     = (float*)d_out;

    const int total   = in_sizes[0];                 // B*C*H*W
    const int batches = total / (kH * kW);           // 32 for the reference shapes

    dim3 grid((unsigned)(batches * (kH / kRows)));   // one block per 8 image rows
    dim3 block(256);                                 // 8 wave32 waves, 4 cols/lane
    hipLaunchKernelGGL(laplacian_rows8_kernel, grid, block, 0, stream, x, out);
}